// WHVILinear_29858612642235
// MI455X (gfx1250) — compile-verified
//
#include <hip/hip_runtime.h>

typedef float v2f __attribute__((ext_vector_type(2)));
typedef float v8f __attribute__((ext_vector_type(8)));

#define DIM   4096
#define BSTR  20            // padded 'b' stride: 16B-aligned float4 rows, conflict-free lane stride
#define ASTR  (16 * BSTR)   // 320
#define LDSN  (16 * ASTR)   // 5120 floats per buffer (20 KB); two buffers = 40 KB

// Sylvester Hadamard sign: H16[i][j] = (-1)^popc(i&j), exact in fp32.
__device__ __forceinline__ float hsign(int i, int j) {
    return (__popc(i & j) & 1) ? -1.0f : 1.0f;
}

__device__ __forceinline__ v8f tile_mm(const v2f* af, const v2f* bf) {
    v8f acc = {};
#pragma unroll
    for (int kc = 0; kc < 4; ++kc)
        acc = __builtin_amdgcn_wmma_f32_16x16x4_f32(false, af[kc], false, bf[kc],
                                                    (short)0, acc, false, false);
    return acc;
}

// Y = X @ H (transform along 'c', tile a = w).
__device__ __forceinline__ void stage_c(const float* __restrict__ src,
                                        float* __restrict__ dst,
                                        const float* hfx, const float* hfy,
                                        int w, int h, int l) {
    const float* sb = src + w * ASTR + l * BSTR + 2 * h;
    float*       db = dst + w * ASTR + 8 * h * BSTR + l;
    v2f af[4], bf[4];
#pragma unroll
    for (int kc = 0; kc < 4; ++kc) {
        af[kc].x = sb[4 * kc];        // A[l][4kc+2h]
        af[kc].y = sb[4 * kc + 1];
        bf[kc].x = hfx[kc];           // H[4kc+2h][l] (= H[l][k] by symmetry)
        bf[kc].y = hfy[kc];
    }
    v8f acc = tile_mm(af, bf);
#pragma unroll
    for (int r = 0; r < 8; ++r) db[r * BSTR] = acc[r];
}

// Z = H @ Y (transform along 'b', tile a = w).
__device__ __forceinline__ void stage_b(const float* __restrict__ src,
                                        float* __restrict__ dst,
                                        const float* hfx, const float* hfy,
                                        int w, int h, int l) {
    const float* sb = src + w * ASTR + 2 * h * BSTR + l;
    float*       db = dst + w * ASTR + 8 * h * BSTR + l;
    v2f af[4], bf[4];
#pragma unroll
    for (int kc = 0; kc < 4; ++kc) {
        af[kc].x = hfx[kc];                    // H[l][4kc+2h]
        af[kc].y = hfy[kc];
        bf[kc].x = sb[(4 * kc) * BSTR];        // Y[4kc+2h][l]
        bf[kc].y = sb[(4 * kc + 1) * BSTR];
    }
    v8f acc = tile_mm(af, bf);
#pragma unroll
    for (int r = 0; r < 8; ++r) db[r * BSTR] = acc[r];
}

// W[:,w,:] = H @ Z[:,w,:] (transform along 'a', tile b = w), output scaled by u[j].
__device__ __forceinline__ void stage_a_u(const float* __restrict__ src,
                                          float* __restrict__ dst,
                                          const float* __restrict__ u,
                                          const float* hfx, const float* hfy,
                                          int w, int h, int l) {
    const float* sb = src + 2 * h * ASTR + w * BSTR + l;
    float*       db = dst + 8 * h * ASTR + w * BSTR + l;
    const float* ub = u + 2048 * h + w * 16 + l;   // j = (r+8h)*256 + w*16 + l
    v2f af[4], bf[4];
#pragma unroll
    for (int kc = 0; kc < 4; ++kc) {
        af[kc].x = hfx[kc];
        af[kc].y = hfy[kc];
        bf[kc].x = sb[(4 * kc) * ASTR];        // Z[4kc+2h][w][l]
        bf[kc].y = sb[(4 * kc + 1) * ASTR];
    }
    v8f acc = tile_mm(af, bf);
#pragma unroll
    for (int r = 0; r < 8; ++r) db[r * ASTR] = acc[r] * ub[r * 256];
}

// Final 'a' transform: scale by s1 and store straight to global (coalesced).
__device__ __forceinline__ void stage_a_out(const float* __restrict__ src,
                                            float* __restrict__ outrow,
                                            const float* __restrict__ s1,
                                            const float* hfx, const float* hfy,
                                            int w, int h, int l) {
    const float* sb  = src + 2 * h * ASTR + w * BSTR + l;
    const float* s1b = s1 + 2048 * h + w * 16 + l;
    float*       ob  = outrow + 2048 * h + w * 16 + l;
    v2f af[4], bf[4];
#pragma unroll
    for (int kc = 0; kc < 4; ++kc) {
        af[kc].x = hfx[kc];
        af[kc].y = hfy[kc];
        bf[kc].x = sb[(4 * kc) * ASTR];
        bf[kc].y = sb[(4 * kc + 1) * ASTR];
    }
    v8f acc = tile_mm(af, bf);
#pragma unroll
    for (int r = 0; r < 8; ++r) ob[r * 256] = acc[r] * s1b[r * 256];
}

// u[j] = g_mu[j] + softplus(g_rho[j]) * epsilon[j]
__global__ void whvi_compute_u(const float* __restrict__ eps,
                               const float* __restrict__ g_mu,
                               const float* __restrict__ g_rho,
                               float* __restrict__ u) {
    const int j = blockIdx.x * blockDim.x + threadIdx.x;
    if (j < DIM) {
        const float sp = log1pf(__expf(g_rho[j]));
        u[j] = g_mu[j] + sp * eps[j];
    }
}

// One workgroup per row: out_row = s1 * FWHT( u * FWHT( s2 * x_row ) )
__global__ __launch_bounds__(512) void whvi_row_kernel(const float* __restrict__ x,
                                                       const float* __restrict__ s1,
                                                       const float* __restrict__ s2,
                                                       const float* __restrict__ u,
                                                       float* __restrict__ out) {
    __shared__ float bufA[LDSN];
    __shared__ float bufB[LDSN];

    const int row  = blockIdx.x;
    const int tid  = threadIdx.x;
    const int wave = tid >> 5;
    const int lane = tid & 31;
    const int h    = lane >> 4;
    const int l    = lane & 15;

    // H16 fragments: computed once, pinned in VGPRs for all 6 stages.
    float hfx[4], hfy[4];
#pragma unroll
    for (int kc = 0; kc < 4; ++kc) {
        const int k0 = 4 * kc + 2 * h;
        hfx[kc] = hsign(l, k0);
        hfy[kc] = hsign(l, k0 + 1);
    }
    asm("" : "+v"(hfx[0]), "+v"(hfx[1]), "+v"(hfx[2]), "+v"(hfx[3]),
             "+v"(hfy[0]), "+v"(hfy[1]), "+v"(hfy[2]), "+v"(hfy[3]));

    // ---- load row, scale by s2, deposit into padded LDS (16B-aligned b128 stores) ----
    const float4* x4  = (const float4*)(x + (size_t)row * DIM);
    const float4* s24 = (const float4*)s2;
#pragma unroll
    for (int t = 0; t < 2; ++t) {
        const int idx4 = tid + t * 512;
        const int j    = idx4 << 2;
        const float4 v = x4[idx4];
        const float4 s = s24[idx4];
        const int p = (j >> 8) * ASTR + ((j >> 4) & 15) * BSTR + (j & 15);
        float4 o;
        o.x = v.x * s.x; o.y = v.y * s.y; o.z = v.z * s.z; o.w = v.w * s.w;
        *(float4*)&bufA[p] = o;
    }
    __syncthreads();

    // ---- FWHT #1 (u-scale fused into last stage) ----
    stage_c  (bufA, bufB, hfx, hfy, wave, h, l); __syncthreads();
    stage_b  (bufB, bufA, hfx, hfy, wave, h, l); __syncthreads();
    stage_a_u(bufA, bufB, u, hfx, hfy, wave, h, l); __syncthreads();

    // ---- FWHT #2 (s1-scale + global store fused into last stage) ----
    stage_c(bufB, bufA, hfx, hfy, wave, h, l); __syncthreads();
    stage_b(bufA, bufB, hfx, hfy, wave, h, l); __syncthreads();
    stage_a_out(bufB, out + (size_t)row * DIM, s1, hfx, hfy, wave, h, l);
}

extern "C" void kernel_launch(void* const* d_in, const int* in_sizes, int n_in,
                              void* d_out, int out_size, void* d_ws, size_t ws_size,
                              hipStream_t stream) {
    const float* x     = (const float*)d_in[0];
    const float* eps   = (const float*)d_in[1];
    const float* s1    = (const float*)d_in[2];
    const float* s2    = (const float*)d_in[3];
    const float* g_mu  = (const float*)d_in[4];
    const float* g_rho = (const float*)d_in[5];
    float* out = (float*)d_out;
    float* u   = (float*)d_ws;   // 4096 floats of scratch

    whvi_compute_u<<<DIM / 256, 256, 0, stream>>>(eps, g_mu, g_rho, u);
    whvi_row_kernel<<<4096, 512, 0, stream>>>(x, s1, s2, u, out);
}